// YOLOv4_23192823398624
// MI455X (gfx1250) — compile-verified
//
#include <hip/hip_runtime.h>
#include <stdint.h>

// ---------------- problem constants (from reference) ----------------
#define NP3 8112
#define NP4 2028
#define NP5 507
#define NTOT (NP3 + NP4 + NP5)     // 10647 boxes per image
#define BATCH 4
#define NCLS 20
#define MSORT 16384                 // next pow2 >= NTOT for bitonic sort
#define CONF_THR 0.1f
#define IOU_THR 0.45f

typedef unsigned int u32;
typedef unsigned int u32x4 __attribute__((ext_vector_type(4)));
typedef int          i32x4 __attribute__((ext_vector_type(4)));
typedef int          i32x8 __attribute__((ext_vector_type(8)));

#if defined(__AMDGCN__)
#  if __has_builtin(__builtin_amdgcn_tensor_load_to_lds)
#    define HAVE_TDM 1
#  endif
#  if __has_builtin(__builtin_amdgcn_s_wait_tensorcnt)
#    define HAVE_TWAIT 1
#  endif
#endif

#ifdef HAVE_TDM
// 1-D bulk DMA of `elems` f32 from global into LDS via the Tensor Data Mover.
// Caller gates to one wave; caller is responsible for the workgroup barrier.
__device__ __forceinline__ void tdm_load_1d(void* lds_dst, const void* gsrc, unsigned elems)
{
    unsigned lds_base = (unsigned)(uintptr_t)lds_dst;          // low 32b = LDS byte offset
    unsigned long long ga = (unsigned long long)(uintptr_t)gsrc;
    u32x4 g0; i32x8 g1; i32x4 z4 = {0, 0, 0, 0};
    g0.x = 1u;                                                 // count=1, is_restore=0
    g0.y = lds_base;                                           // lds_addr (bytes)
    g0.z = (u32)(ga & 0xFFFFFFFFull);                          // global_addr[31:0]
    g0.w = (u32)((ga >> 32) & 0x01FFFFFFull) | (2u << 30);     // addr[56:32] | type=2
    g1[0] = (int)(2u << 16);                                   // data_size=4B; wg_mask=0
    g1[1] = (int)((elems & 0xFFFFu) << 16);                    // tensor_dim0 lo @bit48
    g1[2] = (int)((elems >> 16) | (1u << 16));                 // tensor_dim0 hi | tensor_dim1=1
    g1[3] = (int)((elems & 0xFFFFu) << 16);                    // tile_dim0 @bit112 (<=65535)
    g1[4] = 0;                                                 // tile_dim1/2 unused (1-D)
    g1[5] = (int)elems;                                        // tensor_dim0_stride lo
    g1[6] = 0; g1[7] = 0;
#if __clang_major__ >= 23
    i32x8 z8 = {0, 0, 0, 0, 0, 0, 0, 0};
    __builtin_amdgcn_tensor_load_to_lds(g0, g1, z4, z4, z8, 0);
#else
    __builtin_amdgcn_tensor_load_to_lds(g0, g1, z4, z4, 0);
#endif
#ifdef HAVE_TWAIT
    __builtin_amdgcn_s_wait_tensorcnt(0);
#else
    asm volatile("s_wait_tensorcnt 0x0" ::: "memory");
#endif
}
#endif

// ---------------------------------------------------------------------------
// K1: per-box score / class-argmax / box concat.   grid = B*N lanes.
// score encoding: valid(conf>thr) ? conf*clsmax (>=0) : -1.0  (invalid sinks
// to the tail of the descending sort, in original-index order via tiebreak,
// matching the reference's stable argsort of zero-score invalids).
// ---------------------------------------------------------------------------
__global__ void prep_kernel(const float* __restrict__ p3b, const float* __restrict__ p3c, const float* __restrict__ p3k,
                            const float* __restrict__ p4b, const float* __restrict__ p4c, const float* __restrict__ p4k,
                            const float* __restrict__ p5b, const float* __restrict__ p5c, const float* __restrict__ p5k,
                            float4* __restrict__ boxesc, float* __restrict__ score, int* __restrict__ clsix)
{
    int gid = blockIdx.x * blockDim.x + threadIdx.x;
    if (gid >= BATCH * NTOT) return;
    int b = gid / NTOT;
    int n = gid - b * NTOT;

    const float *bb, *cc, *kk;
    int i, len;
    if (n < NP3)             { bb = p3b; cc = p3c; kk = p3k; i = n;             len = NP3; }
    else if (n < NP3 + NP4)  { bb = p4b; cc = p4c; kk = p4k; i = n - NP3;       len = NP4; }
    else                     { bb = p5b; cc = p5c; kk = p5k; i = n - NP3 - NP4; len = NP5; }

    int src = b * len + i;
    const float4* cr4 = (const float4*)(kk + (size_t)src * NCLS);  // 80B row, 16B aligned
    __builtin_prefetch(cr4, 0, 0);                 // global_prefetch_b8 on the class row

    float4 bx   = ((const float4*)bb)[src];
    float  conf = cc[src];

    // first-occurrence argmax over 20 classes, 5x b128 loads
    float m = -1.0f; int mi = 0;
    #pragma unroll
    for (int k = 0; k < NCLS / 4; ++k) {
        float4 q = cr4[k];
        if (q.x > m) { m = q.x; mi = 4 * k + 0; }
        if (q.y > m) { m = q.y; mi = 4 * k + 1; }
        if (q.z > m) { m = q.z; mi = 4 * k + 2; }
        if (q.w > m) { m = q.w; mi = 4 * k + 3; }
    }

    boxesc[gid] = bx;
    score[gid]  = (conf > CONF_THR) ? conf * m : -1.0f;
    clsix[gid]  = mi;
}

// ---------------------------------------------------------------------------
// K2: full in-LDS bitonic sort of 16384 (key,idx) slots per image.
// 128 KB LDS per workgroup -- fits CDNA5's 320 KB WGP LDS, one WG per image.
// Score vector is DMA'd into the key region by the TDM while other waves
// initialize indices and padding (disjoint LDS ranges).
// ---------------------------------------------------------------------------
__global__ void __launch_bounds__(1024)
sort_kernel(const float4* __restrict__ boxesc, const float* __restrict__ score,
            const int* __restrict__ clsix, float* __restrict__ sdata,
            float* __restrict__ out)
{
    extern __shared__ char smem[];
    float* skey = (float*)smem;
    int*   sidx = (int*)(smem + (size_t)MSORT * sizeof(float));
    int b = blockIdx.x, tid = threadIdx.x;

#ifdef HAVE_TDM
    if (tid < 32) tdm_load_1d(skey, score + (size_t)b * NTOT, NTOT);
    for (int m = tid; m < MSORT; m += 1024) {
        if (m >= NTOT) skey[m] = -2.0f;            // padding below any real key
        sidx[m] = m;
    }
#else
    for (int m = tid; m < MSORT; m += 1024) {
        skey[m] = (m < NTOT) ? score[b * NTOT + m] : -2.0f;
        sidx[m] = m;
    }
#endif
    __syncthreads();

    for (int k = 2; k <= MSORT; k <<= 1) {
        for (int j = k >> 1; j > 0; j >>= 1) {
            for (int t = tid; t < MSORT / 2; t += 1024) {
                int i = ((t & ~(j - 1)) << 1) | (t & (j - 1));
                int p = i | j;
                float a = skey[i], c = skey[p];
                int  ia = sidx[i], ic = sidx[p];
                bool aBeforeC   = (a > c) || (a == c && ia < ic);   // desired order
                bool descRegion = ((i & k) == 0);
                if (descRegion ? !aBeforeC : aBeforeC) {
                    skey[i] = c; skey[p] = a; sidx[i] = ic; sidx[p] = ia;
                }
            }
            __syncthreads();
        }
    }

    for (int m = tid; m < NTOT; m += 1024) {
        float key = skey[m];
        int   id  = sidx[m];
        float4 bx = boxesc[b * NTOT + id];
        float area = (bx.z - bx.x) * (bx.w - bx.y);
        size_t so = ((size_t)b * NTOT + m) * 6;
        sdata[so + 0] = bx.x; sdata[so + 1] = bx.y;
        sdata[so + 2] = bx.z; sdata[so + 3] = bx.w;
        sdata[so + 4] = area; sdata[so + 5] = key;     // key>=0 <=> valid
        out[so + 0] = bx.x; out[so + 1] = bx.y;
        out[so + 2] = bx.z; out[so + 3] = bx.w;
        out[so + 5] = (float)clsix[b * NTOT + id];     // col 4 filled by NMS kernel
    }
}

// ---------------------------------------------------------------------------
// K3: greedy NMS fully resident in LDS. One WG (32 wave32 waves) per image.
// LDS: 63,882 f32 tile (249.5 KB, one TDM descriptor) + 10,647-byte
// suppression array => ~260 KB < 320 KB.
// IoU test is division-free: inter/(u) > t  <=>  inter*(1+t) > t*(ar_i+ar_j).
// ---------------------------------------------------------------------------
__global__ void __launch_bounds__(1024)
nms_kernel(const float* __restrict__ sdata, float* __restrict__ out)
{
    extern __shared__ char smem[];
    float* L = (float*)smem;                                   // [NTOT*6]
    unsigned char* suppb = (unsigned char*)(smem + (size_t)NTOT * 6 * sizeof(float));
    int b = blockIdx.x, tid = threadIdx.x;
    const float* gsrc = sdata + (size_t)b * NTOT * 6;

#ifdef HAVE_TDM
    if (tid < 32) tdm_load_1d(L, gsrc, NTOT * 6);              // 63882 f32 in one DMA
#else
    for (int m = tid; m < NTOT * 6; m += 1024) L[m] = gsrc[m];
#endif
    for (int m = tid; m < NTOT; m += 1024) suppb[m] = 0;
    __syncthreads();

    const float tA = 1.0f + IOU_THR;     // 1.45
    const float tB = IOU_THR;            // 0.45

    // Greedy scan: pivot i broadcast from LDS; 1024-lane IoU sweep over j>i.
    for (int i = 0; i < NTOT; ++i) {
        const float* Li = L + i * 6;
        float raw = Li[5];
        if (raw >= 0.0f && suppb[i] == 0) {       // valid and unsuppressed pivot
            float2 pv0 = *(const float2*)(Li + 0);    // x1,y1
            float2 pv1 = *(const float2*)(Li + 2);    // x2,y2
            float  ar  = Li[4];
            for (int j = i + 1 + tid; j < NTOT; j += 1024) {
                const float* Lj = L + j * 6;          // 24B stride, 8B aligned
                float2 q0 = *(const float2*)(Lj + 0); // 3x ds_load_b64
                float2 q1 = *(const float2*)(Lj + 2);
                float2 q2 = *(const float2*)(Lj + 4);
                float iw = fminf(pv1.x, q1.x) - fmaxf(pv0.x, q0.x);
                float ih = fminf(pv1.y, q1.y) - fmaxf(pv0.y, q0.y);
                float inter = fmaxf(iw, 0.0f) * fmaxf(ih, 0.0f);
                if (inter * tA > tB * (ar + q2.x)) suppb[j] = 1;  // iou > IOU_THR
            }
        }
        __syncthreads();
    }

    // Emit keep mask and dets column 4 (score where kept, else 0).
    for (int m = tid; m < NTOT; m += 1024) {
        float raw = L[m * 6 + 5];
        bool  kp  = (raw >= 0.0f) && (suppb[m] == 0);
        out[((size_t)b * NTOT + m) * 6 + 4] = kp ? fmaxf(raw, 0.0f) : 0.0f;
        out[(size_t)BATCH * NTOT * 6 + (size_t)b * NTOT + m] = kp ? 1.0f : 0.0f;
    }
}

// ---------------------------------------------------------------------------
extern "C" void kernel_launch(void* const* d_in, const int* in_sizes, int n_in,
                              void* d_out, int out_size, void* d_ws, size_t ws_size,
                              hipStream_t stream)
{
    const float* p3b = (const float*)d_in[0];
    const float* p3c = (const float*)d_in[1];
    const float* p3k = (const float*)d_in[2];
    const float* p4b = (const float*)d_in[3];
    const float* p4c = (const float*)d_in[4];
    const float* p4k = (const float*)d_in[5];
    const float* p5b = (const float*)d_in[6];
    const float* p5c = (const float*)d_in[7];
    const float* p5k = (const float*)d_in[8];

    char* ws = (char*)d_ws;
    size_t off = 0;
    float4* boxesc = (float4*)(ws + off); off += (size_t)BATCH * NTOT * sizeof(float4);
    float*  score  = (float*)(ws + off);  off += (size_t)BATCH * NTOT * sizeof(float);
    int*    clsix  = (int*)(ws + off);    off += (size_t)BATCH * NTOT * sizeof(int);
    off = (off + 255) & ~(size_t)255;
    float*  sdata  = (float*)(ws + off);  // [B, NTOT, 6] sorted tile (~1 MB)

    float* out = (float*)d_out;

    int total = BATCH * NTOT;
    prep_kernel<<<(total + 255) / 256, 256, 0, stream>>>(
        p3b, p3c, p3k, p4b, p4c, p4k, p5b, p5c, p5k, boxesc, score, clsix);

    sort_kernel<<<BATCH, 1024, (size_t)MSORT * 8, stream>>>(
        boxesc, score, clsix, sdata, out);

    size_t nms_lds = (size_t)NTOT * 6 * sizeof(float) + (((size_t)NTOT + 15) & ~(size_t)15);
    nms_kernel<<<BATCH, 1024, nms_lds, stream>>>(sdata, out);
}